// GATWithAttentionPool_39238821216990
// MI455X (gfx1250) — compile-verified
//
#include <hip/hip_runtime.h>
#include <math.h>

#define NN    50000
#define EE    800000
#define EP    850000        /* EE + NN self-loops */
#define IND   128
#define HIDD  96
#define HEADS 4
#define DHD   24
#define GG    64
#define OUTD  3

typedef __attribute__((ext_vector_type(2))) float v2f;
typedef __attribute__((ext_vector_type(8))) float v8f;

// ---- monotone float<->u32 map for atomic max on floats ----
__device__ __forceinline__ unsigned fmap_u(float f) {
  unsigned u = __float_as_uint(f);
  return (u & 0x80000000u) ? ~u : (u | 0x80000000u);
}
__device__ __forceinline__ float funmap_u(unsigned u) {
  unsigned b = (u & 0x80000000u) ? (u ^ 0x80000000u) : ~u;
  return __uint_as_float(b);
}

// =======================================================================
// GEMM via V_WMMA_F32_16X16X4_F32:  H[rows x 96] = X[rows x K] @ W[K x 96]
// rows % 16 == 0, K % 4 == 0. One wave per 16x16 output tile.
// A 16x4 f32 layout: lane%16 = M, (lane/16)*2+vgpr = K.  B mirrored.
// D: vgpr v -> M = v + 8*(lane/16), N = lane%16.
// =======================================================================
__global__ void gemm_wmma_f32(const float* __restrict__ X,
                              const float* __restrict__ W,
                              float* __restrict__ H,
                              int rows, int K) {
  const int lane   = threadIdx.x & 31;
  const int wave   = threadIdx.x >> 5;
  const int tile_m = blockIdx.x * (blockDim.x >> 5) + wave;
  const int tile_n = blockIdx.y;
  if (tile_m >= (rows >> 4)) return;        // whole-wave uniform branch

  const int row   = tile_m * 16 + (lane & 15);
  const int col   = tile_n * 16 + (lane & 15);
  const int khalf = (lane >> 4) * 2;        // 0 or 2

  v8f c = {};
  for (int k0 = 0; k0 < K; k0 += 4) {
    const int kb = k0 + khalf;
    v2f a, b;
    a[0] = X[(size_t)row * K + kb];
    a[1] = X[(size_t)row * K + kb + 1];
    b[0] = W[(size_t)kb * HIDD + col];
    b[1] = W[(size_t)(kb + 1) * HIDD + col];
    c = __builtin_amdgcn_wmma_f32_16x16x4_f32(false, a, false, b,
                                              (short)0, c, false, false);
  }
  const int rbase = tile_m * 16 + ((lane >> 4) << 3);
#pragma unroll
  for (int v = 0; v < 8; ++v)
    H[(size_t)(rbase + v) * HIDD + col] = c[v];
}

// ---------------- utility fill ----------------
__global__ void fill_u32(unsigned* __restrict__ p, unsigned v, int n) {
  int i = blockIdx.x * blockDim.x + threadIdx.x;
  if (i < n) p[i] = v;
}

// ---------------- per-node attention logits ----------------
__global__ void alpha_kernel(const float* __restrict__ h,
                             const float* __restrict__ a_src,
                             const float* __restrict__ a_dst,
                             float* __restrict__ als, float* __restrict__ ald) {
  int i = blockIdx.x * blockDim.x + threadIdx.x;   // n*HEADS + head
  if (i >= NN * HEADS) return;
  int n = i / HEADS, hh = i % HEADS;
  const float* hp  = h + (size_t)n * HIDD + hh * DHD;
  const float* asp = a_src + hh * DHD;
  const float* adp = a_dst + hh * DHD;
  float ss = 0.f, sd = 0.f;
#pragma unroll
  for (int d = 0; d < DHD; ++d) { float v = hp[d]; ss += v * asp[d]; sd += v * adp[d]; }
  als[i] = ss; ald[i] = sd;
}

__device__ __forceinline__ void edge_ends(const int* __restrict__ ei, int e,
                                          int& s, int& d) {
  if (e < EE) { s = ei[e]; d = ei[EE + e]; }
  else        { s = e - EE; d = s; }               // self loop
}

// ---------------- segment max of leaky-relu logits ----------------
__global__ void edge_max_kernel(const int* __restrict__ ei,
                                const float* __restrict__ als,
                                const float* __restrict__ ald,
                                unsigned* __restrict__ mx) {
  int e = blockIdx.x * blockDim.x + threadIdx.x;
  if (e >= EP) return;
  int s, d; edge_ends(ei, e, s, d);
#pragma unroll
  for (int hh = 0; hh < HEADS; ++hh) {
    float v = als[s * HEADS + hh] + ald[d * HEADS + hh];
    v = v > 0.f ? v : 0.2f * v;
    atomicMax(&mx[d * HEADS + hh], fmap_u(v));
  }
}

// ---------------- segment sum of exp ----------------
__global__ void edge_expsum_kernel(const int* __restrict__ ei,
                                   const float* __restrict__ als,
                                   const float* __restrict__ ald,
                                   const unsigned* __restrict__ mx,
                                   float* __restrict__ den) {
  int e = blockIdx.x * blockDim.x + threadIdx.x;
  if (e >= EP) return;
  int s, d; edge_ends(ei, e, s, d);
#pragma unroll
  for (int hh = 0; hh < HEADS; ++hh) {
    float v = als[s * HEADS + hh] + ald[d * HEADS + hh];
    v = v > 0.f ? v : 0.2f * v;
    atomicAdd(&den[d * HEADS + hh], __expf(v - funmap_u(mx[d * HEADS + hh])));
  }
}

// ---------------- weighted message scatter: one wave per edge ----------------
__global__ void edge_agg_kernel(const int* __restrict__ ei,
                                const float* __restrict__ hfeat,
                                const float* __restrict__ als,
                                const float* __restrict__ ald,
                                const unsigned* __restrict__ mx,
                                const float* __restrict__ den,
                                float* __restrict__ agg) {
  const int lane = threadIdx.x & 31;
  const int e    = blockIdx.x * (blockDim.x >> 5) + (threadIdx.x >> 5);
  if (e >= EP) return;
  int s, d; edge_ends(ei, e, s, d);
#pragma unroll
  for (int it = 0; it < 3; ++it) {               // 96 = 3*32 channels
    int c  = it * 32 + lane;
    int hh = c / DHD;
    float v = als[s * HEADS + hh] + ald[d * HEADS + hh];
    v = v > 0.f ? v : 0.2f * v;
    float ex    = __expf(v - funmap_u(mx[d * HEADS + hh]));
    float alpha = ex / (den[d * HEADS + hh] + 1e-16f);
    atomicAdd(&agg[(size_t)d * HIDD + c], hfeat[(size_t)s * HIDD + c] * alpha);
  }
}

// ---------------- bias + ELU in place ----------------
__global__ void bias_elu_kernel(float* __restrict__ agg, const float* __restrict__ b) {
  int i = blockIdx.x * blockDim.x + threadIdx.x;
  if (i >= NN * HIDD) return;
  float v = agg[i] + b[i % HIDD];
  agg[i] = v > 0.f ? v : (__expf(v) - 1.f);
}

// ---------------- gate MLP per node + per-graph max ----------------
__global__ void gate_kernel(const float* __restrict__ h,
                            const float* __restrict__ gw1, const float* __restrict__ gb1,
                            const float* __restrict__ gw2, const float* __restrict__ gb2,
                            const int* __restrict__ batch,
                            float* __restrict__ gate, unsigned* __restrict__ gmx) {
  int n = blockIdx.x * blockDim.x + threadIdx.x;
  if (n >= NN) return;
  const float* hp = h + (size_t)n * HIDD;
  float g = gb2[0];
  for (int j = 0; j < HIDD / 2; ++j) {
    float s = gb1[j];
    for (int k = 0; k < HIDD; ++k) s += hp[k] * gw1[k * (HIDD / 2) + j];
    g += (s > 0.f ? s : 0.f) * gw2[j];
  }
  gate[n] = g;
  atomicMax(&gmx[batch[n]], fmap_u(g));
}

__global__ void gfix_kernel(const unsigned* __restrict__ gmx, float* __restrict__ gm) {
  int g = threadIdx.x;
  if (g < GG) {
    float v = funmap_u(gmx[g]);
    gm[g] = (fabsf(v) <= 3.4e38f) ? v : 0.f;     // isfinite -> else 0 (empty graph)
  }
}

__global__ void gden_kernel(const float* __restrict__ gate, const int* __restrict__ batch,
                            const float* __restrict__ gm, float* __restrict__ gden) {
  int n = blockIdx.x * blockDim.x + threadIdx.x;
  if (n >= NN) return;
  int b = batch[n];
  atomicAdd(&gden[b], __expf(gate[n] - gm[b]));
}

__global__ void pool_kernel(const float* __restrict__ h, const float* __restrict__ gate,
                            const int* __restrict__ batch, const float* __restrict__ gm,
                            const float* __restrict__ gden, float* __restrict__ pooled) {
  int i = blockIdx.x * blockDim.x + threadIdx.x;
  if (i >= NN * HIDD) return;
  int n = i / HIDD, c = i % HIDD;
  int b = batch[n];
  float w = __expf(gate[n] - gm[b]) / (gden[b] + 1e-16f);
  atomicAdd(&pooled[b * HIDD + c], h[i] * w);
}

// ---------------- final MLP over 64 graphs ----------------
__global__ void final_kernel(const float* __restrict__ pooled,
                             const float* __restrict__ fw1, const float* __restrict__ fb1,
                             const float* __restrict__ fw2, const float* __restrict__ fb2,
                             float* __restrict__ out) {
  __shared__ float t[HIDD];
  int g = blockIdx.x, c = threadIdx.x;
  const float* pp = pooled + g * HIDD;
  float s = fb1[c];
  for (int k = 0; k < HIDD; ++k) s += pp[k] * fw1[k * HIDD + c];
  t[c] = s > 0.f ? s : 0.f;
  __syncthreads();
  if (c < OUTD) {
    float o = fb2[c];
    for (int k = 0; k < HIDD; ++k) o += t[k] * fw2[k * OUTD + c];
    out[g * OUTD + c] = o;
  }
}

extern "C" void kernel_launch(void* const* d_in, const int* in_sizes, int n_in,
                              void* d_out, int out_size, void* d_ws, size_t ws_size,
                              hipStream_t stream) {
  const float* x   = (const float*)d_in[0];
  const int*   ei  = (const int*)d_in[1];
  const int*   bat = (const int*)d_in[2];
  const float* W1  = (const float*)d_in[4];
  const float* at_s1 = (const float*)d_in[5];
  const float* at_d1 = (const float*)d_in[6];
  const float* b1  = (const float*)d_in[7];
  const float* W2  = (const float*)d_in[8];
  const float* at_s2 = (const float*)d_in[9];
  const float* at_d2 = (const float*)d_in[10];
  const float* b2  = (const float*)d_in[11];
  const float* gw1 = (const float*)d_in[12];
  const float* gb1 = (const float*)d_in[13];
  const float* gw2 = (const float*)d_in[14];
  const float* gb2 = (const float*)d_in[15];
  const float* fw1 = (const float*)d_in[16];
  const float* fb1 = (const float*)d_in[17];
  const float* fw2 = (const float*)d_in[18];
  const float* fb2 = (const float*)d_in[19];
  float* out = (float*)d_out;

  // workspace carve-up (floats)
  float*    ws    = (float*)d_ws;
  float*    buf0  = ws;                                 // N*96  (transform h)
  float*    buf1  = buf0 + (size_t)NN * HIDD;           // N*96  (agg / activated)
  float*    als   = buf1 + (size_t)NN * HIDD;           // N*4
  float*    ald   = als + NN * HEADS;                   // N*4
  unsigned* mx    = (unsigned*)(ald + NN * HEADS);      // N*4
  float*    den   = (float*)(mx + NN * HEADS);          // N*4
  float*    gate  = den + NN * HEADS;                   // N
  unsigned* gmx   = (unsigned*)(gate + NN);             // G
  float*    gm    = (float*)(gmx + GG);                 // G
  float*    gden  = gm + GG;                            // G
  float*    pooled= gden + GG;                          // G*96

  const int T = 256;
  auto cdiv = [](int a, int b) { return (a + b - 1) / b; };
  const dim3 ggrid(cdiv(NN / 16, 8), HIDD / 16);        // 8 waves/block, 6 col tiles

  // ================= Layer 1 =================
  gemm_wmma_f32<<<ggrid, 256, 0, stream>>>(x, W1, buf0, NN, IND);
  fill_u32<<<cdiv(NN * HIDD, T), T, 0, stream>>>((unsigned*)buf1, 0u, NN * HIDD);
  fill_u32<<<cdiv(NN * HEADS, T), T, 0, stream>>>(mx, 0u, NN * HEADS);
  fill_u32<<<cdiv(NN * HEADS, T), T, 0, stream>>>((unsigned*)den, 0u, NN * HEADS);
  alpha_kernel<<<cdiv(NN * HEADS, T), T, 0, stream>>>(buf0, at_s1, at_d1, als, ald);
  edge_max_kernel<<<cdiv(EP, T), T, 0, stream>>>(ei, als, ald, mx);
  edge_expsum_kernel<<<cdiv(EP, T), T, 0, stream>>>(ei, als, ald, mx, den);
  edge_agg_kernel<<<cdiv(EP, 8), 256, 0, stream>>>(ei, buf0, als, ald, mx, den, buf1);
  bias_elu_kernel<<<cdiv(NN * HIDD, T), T, 0, stream>>>(buf1, b1);

  // ================= Layer 2 =================
  gemm_wmma_f32<<<ggrid, 256, 0, stream>>>(buf1, W2, buf0, NN, HIDD);
  fill_u32<<<cdiv(NN * HIDD, T), T, 0, stream>>>((unsigned*)buf1, 0u, NN * HIDD);
  fill_u32<<<cdiv(NN * HEADS, T), T, 0, stream>>>(mx, 0u, NN * HEADS);
  fill_u32<<<cdiv(NN * HEADS, T), T, 0, stream>>>((unsigned*)den, 0u, NN * HEADS);
  alpha_kernel<<<cdiv(NN * HEADS, T), T, 0, stream>>>(buf0, at_s2, at_d2, als, ald);
  edge_max_kernel<<<cdiv(EP, T), T, 0, stream>>>(ei, als, ald, mx);
  edge_expsum_kernel<<<cdiv(EP, T), T, 0, stream>>>(ei, als, ald, mx, den);
  edge_agg_kernel<<<cdiv(EP, 8), 256, 0, stream>>>(ei, buf0, als, ald, mx, den, buf1);
  bias_elu_kernel<<<cdiv(NN * HIDD, T), T, 0, stream>>>(buf1, b2);

  // ================= Global attention pool + head =================
  fill_u32<<<1, GG, 0, stream>>>(gmx, 0x007FFFFFu, GG);     // map(-inf)
  fill_u32<<<1, GG, 0, stream>>>((unsigned*)gden, 0u, GG);
  fill_u32<<<cdiv(GG * HIDD, T), T, 0, stream>>>((unsigned*)pooled, 0u, GG * HIDD);
  gate_kernel<<<cdiv(NN, T), T, 0, stream>>>(buf1, gw1, gb1, gw2, gb2, bat, gate, gmx);
  gfix_kernel<<<1, GG, 0, stream>>>(gmx, gm);
  gden_kernel<<<cdiv(NN, T), T, 0, stream>>>(gate, bat, gm, gden);
  pool_kernel<<<cdiv(NN * HIDD, T), T, 0, stream>>>(buf1, gate, bat, gm, gden, pooled);
  final_kernel<<<GG, HIDD, 0, stream>>>(pooled, fw1, fb1, fw2, fb2, out);
}